// GCNDrugEncoder_71734543777906
// MI455X (gfx1250) — compile-verified
//
#include <hip/hip_runtime.h>

#define NNODES 200000
#define NEDGES 400000
#define HD     128
#define NLAYERS 3
#define NGRAPH 4096
#define BN_EPS 1e-5f

typedef __attribute__((ext_vector_type(2))) float v2f;
typedef __attribute__((ext_vector_type(8))) float v8f;

// ---------------- degree / norm ----------------
__global__ void k_init_deg(float* deg) {
    int i = blockIdx.x * blockDim.x + threadIdx.x;
    if (i < NNODES) deg[i] = 1.0f;              // self-loop
}
__global__ void k_edge_deg(const int* __restrict__ col, float* __restrict__ deg) {
    int i = blockIdx.x * blockDim.x + threadIdx.x;
    if (i < NEDGES) atomicAdd(&deg[col[i]], 1.0f);
}
__global__ void k_rsqrt(float* deg) {
    int i = blockIdx.x * blockDim.x + threadIdx.x;
    if (i < NNODES) deg[i] = rsqrtf(deg[i]);    // deg >= 1 always
}

// ---------------- WMMA f32 GEMM: C[rows x 128] = scale(A) @ W + bias ---------
// block = 256 threads = 8 waves; block computes 64 rows x 128 cols.
// wave w owns columns [16w,16w+16) and all 4 row-tiles (4 independent v8f accs).
// K processed in 2 phases of 64; per phase: A chunk (64x64, 16KB) and the
// K-pair-interleaved W chunk (32x256, 32KB) are staged in LDS so every WMMA
// operand fragment is a single ds_load_b64.
__launch_bounds__(256)
__global__ void k_gemm_wmma(const float* __restrict__ A,
                            const float* __restrict__ W,
                            const float* __restrict__ bias,      // may be null
                            const float* __restrict__ rowscale,  // may be null
                            float* __restrict__ C) {
    __shared__ float As[64 * 64];    // 16 KB: As[r][kk2] = A[row0+r][k0+kk2] (scaled)
    __shared__ float Ws[32 * 256];   // 32 KB: Ws[t][n*2+j] = W[k0+2t+j][n]
    const int row0 = blockIdx.x * 64;

    const int lane = threadIdx.x & 31;
    const int wv   = threadIdx.x >> 5;            // 0..7 -> column tile
    const int mA   = lane & 15;                   // row within 16-row tile
    const int kkL  = (lane >> 4) << 1;            // lane K sub-pair: 0 or 2
    const int nB   = wv * 16 + (lane & 15);       // B/C column (0..127)

    v8f acc0 = {}, acc1 = {}, acc2 = {}, acc3 = {};

    #pragma unroll
    for (int p = 0; p < 2; ++p) {
        const int k0 = p * 64;
        // ---- stage A chunk: 64 rows x 64 cols, float4 per item ----
        #pragma unroll
        for (int it = 0; it < 4; ++it) {
            int w = it * 256 + threadIdx.x;       // 0..1023
            int r = w >> 4, c4 = w & 15;          // 16 float4 per row
            float4 v = *(const float4*)&A[(size_t)(row0 + r) * HD + k0 + c4 * 4];
            if (rowscale) {
                float s = rowscale[row0 + r];
                v.x *= s; v.y *= s; v.z *= s; v.w *= s;
            }
            *(float4*)&As[r * 64 + c4 * 4] = v;
        }
        // ---- stage W chunk, K-pair interleaved ----
        #pragma unroll
        for (int it = 0; it < 4; ++it) {
            int w = it * 256 + threadIdx.x;       // 0..1023
            int t = w >> 5, n4 = w & 31;          // t: K-pair 0..31, n4: col group
            const float* w0 = &W[(size_t)(k0 + 2 * t) * HD + n4 * 4];
            float4 r0 = *(const float4*)w0;
            float4 r1 = *(const float4*)(w0 + HD);
            float4 lo = {r0.x, r1.x, r0.y, r1.y};
            float4 hi = {r0.z, r1.z, r0.w, r1.w};
            float* dst = &Ws[t * 256 + n4 * 8];
            *(float4*)dst       = lo;
            *(float4*)(dst + 4) = hi;
        }
        __syncthreads();

        // ---- 16 K-steps of 4: 1 B frag + 4 A frags -> 4 WMMAs ----
        #pragma unroll
        for (int kk2 = 0; kk2 < 64; kk2 += 4) {
            const int tt = (kk2 + kkL) >> 1;
            v2f b  = *(const v2f*)&Ws[tt * 256 + nB * 2];
            v2f a0 = *(const v2f*)&As[(0 * 16 + mA) * 64 + kk2 + kkL];
            v2f a1 = *(const v2f*)&As[(1 * 16 + mA) * 64 + kk2 + kkL];
            v2f a2 = *(const v2f*)&As[(2 * 16 + mA) * 64 + kk2 + kkL];
            v2f a3 = *(const v2f*)&As[(3 * 16 + mA) * 64 + kk2 + kkL];
            acc0 = __builtin_amdgcn_wmma_f32_16x16x4_f32(false, a0, false, b, (short)0, acc0, false, false);
            acc1 = __builtin_amdgcn_wmma_f32_16x16x4_f32(false, a1, false, b, (short)0, acc1, false, false);
            acc2 = __builtin_amdgcn_wmma_f32_16x16x4_f32(false, a2, false, b, (short)0, acc2, false, false);
            acc3 = __builtin_amdgcn_wmma_f32_16x16x4_f32(false, a3, false, b, (short)0, acc3, false, false);
        }
        __syncthreads();
    }

    const int mtop = (lane >> 4) * 8;             // lanes16-31 hold M = v+8
    const float bn = bias ? bias[nB] : 0.0f;
    #pragma unroll
    for (int v = 0; v < 8; ++v) {
        C[(size_t)(row0 +  0 + mtop + v) * HD + nB] = acc0[v] + bn;
        C[(size_t)(row0 + 16 + mtop + v) * HD + nB] = acc1[v] + bn;
        C[(size_t)(row0 + 32 + mtop + v) * HD + nB] = acc2[v] + bn;
        C[(size_t)(row0 + 48 + mtop + v) * HD + nB] = acc3[v] + bn;
    }
}

// ------------- hn = b_conv + hm * dinv^2 (self-loop term), float4/thread ----
__global__ void k_msg_init(const float4* __restrict__ hm4,
                           const float* __restrict__ dinv,
                           const float* __restrict__ bconv,
                           float4* __restrict__ hn4) {
    size_t idx = (size_t)blockIdx.x * 256 + threadIdx.x;   // N*32 threads
    int row = (int)(idx >> 5), f4 = (int)(idx & 31);
    float w = dinv[row]; w *= w;
    float4 m = hm4[idx];
    const float4 b = ((const float4*)bconv)[f4];
    float4 r;
    r.x = b.x + m.x * w; r.y = b.y + m.y * w;
    r.z = b.z + m.z * w; r.w = b.w + m.w * w;
    hn4[idx] = r;
}

// ------------- edge scatter: hn[col] += hm[row] * dinv[row]*dinv[col] -------
// one wave (32 lanes x float4) per edge
__global__ void k_scatter(const float4* __restrict__ hm4,
                          const int* __restrict__ row,
                          const int* __restrict__ col,
                          const float* __restrict__ dinv,
                          float* __restrict__ hn) {
    size_t gid = (size_t)blockIdx.x * 256 + threadIdx.x;   // E*32 threads
    int e = (int)(gid >> 5), lane = (int)(gid & 31);
    int r = row[e], c = col[e];
    float w = dinv[r] * dinv[c];
    float4 m = hm4[(size_t)r * 32 + lane];
    float* dst = hn + (size_t)c * HD + lane * 4;
    atomicAdd(dst + 0, m.x * w);
    atomicAdd(dst + 1, m.y * w);
    atomicAdd(dst + 2, m.z * w);
    atomicAdd(dst + 3, m.w * w);
}

// ------------- BatchNorm statistics: sums[0:128]=sum, sums[128:256]=sumsq ---
__global__ void k_bn_stats(const float* __restrict__ hn, float* __restrict__ sums) {
    int f = threadIdx.x & 127;
    int half = threadIdx.x >> 7;
    float s = 0.f, ss = 0.f;
    for (int i = blockIdx.x * 2 + half; i < NNODES; i += gridDim.x * 2) {
        float v = hn[(size_t)i * HD + f];
        s += v; ss += v * v;
    }
    __shared__ float sh[512];
    sh[threadIdx.x] = s; sh[256 + threadIdx.x] = ss;
    __syncthreads();
    if (threadIdx.x < 128) {
        atomicAdd(&sums[f],       sh[threadIdx.x] + sh[threadIdx.x + 128]);
        atomicAdd(&sums[128 + f], sh[256 + threadIdx.x] + sh[256 + threadIdx.x + 128]);
    }
}

// ------------- BN apply + ReLU + residual: h += relu(norm(hn)) --------------
__global__ void k_bn_apply(const float4* __restrict__ hn4,
                           const float* __restrict__ sums,
                           const float* __restrict__ gamma,
                           const float* __restrict__ beta,
                           float4* __restrict__ h4) {
    size_t idx = (size_t)blockIdx.x * 256 + threadIdx.x;   // N*32 threads
    int f = (int)(idx & 31) * 4;
    float4 v = hn4[idx];
    float4 hv = h4[idx];
    const float invN = 1.0f / (float)NNODES;
    float vv[4] = {v.x, v.y, v.z, v.w};
    float hh[4] = {hv.x, hv.y, hv.z, hv.w};
    #pragma unroll
    for (int j = 0; j < 4; ++j) {
        float mu  = sums[f + j] * invN;
        float var = sums[128 + f + j] * invN - mu * mu;
        float val = (vv[j] - mu) * rsqrtf(var + BN_EPS) * gamma[f + j] + beta[f + j];
        hh[j] += fmaxf(val, 0.0f);
    }
    hv.x = hh[0]; hv.y = hh[1]; hv.z = hh[2]; hv.w = hh[3];
    h4[idx] = hv;
}

// ------------- pooling: gsum[batch[i]] += h[i]; gcnt[batch[i]] += 1 ---------
__global__ void k_pool(const float4* __restrict__ h4,
                       const int* __restrict__ batch,
                       float* __restrict__ gsum,
                       float* __restrict__ gcnt) {
    size_t idx = (size_t)blockIdx.x * 256 + threadIdx.x;   // N*32 threads
    int rowi = (int)(idx >> 5), f4 = (int)(idx & 31);
    int b = batch[rowi];
    float4 v = h4[idx];
    float* dst = gsum + (size_t)b * HD + f4 * 4;
    atomicAdd(dst + 0, v.x);
    atomicAdd(dst + 1, v.y);
    atomicAdd(dst + 2, v.z);
    atomicAdd(dst + 3, v.w);
    if (f4 == 0) atomicAdd(&gcnt[b], 1.0f);
}

__global__ void k_invcnt(float* gcnt) {
    int i = blockIdx.x * blockDim.x + threadIdx.x;
    if (i < NGRAPH) gcnt[i] = 1.0f / fmaxf(gcnt[i], 1.0f);
}

__global__ void k_zero(float* p, int n) {
    int i = blockIdx.x * blockDim.x + threadIdx.x;
    if (i < n) p[i] = 0.0f;
}

extern "C" void kernel_launch(void* const* d_in, const int* in_sizes, int n_in,
                              void* d_out, int out_size, void* d_ws, size_t ws_size,
                              hipStream_t stream) {
    const float* x      = (const float*)d_in[0];
    const float* W_in   = (const float*)d_in[1];
    const float* b_in   = (const float*)d_in[2];
    const float* W_conv = (const float*)d_in[3];
    const float* b_conv = (const float*)d_in[4];
    const float* gamma  = (const float*)d_in[5];
    const float* beta   = (const float*)d_in[6];
    const float* W_out  = (const float*)d_in[7];
    const float* b_out  = (const float*)d_in[8];
    const int*   edge   = (const int*)d_in[9];     // [2, E]: row then col
    const int*   batch  = (const int*)d_in[10];
    float* out = (float*)d_out;

    const int* erow = edge;
    const int* ecol = edge + NEDGES;

    // workspace carve-up (float units, 256-float aligned)
    float* ws = (float*)d_ws;
    size_t o = 0;
    float* dinv = ws + o; o += ((size_t)NNODES + 255) & ~(size_t)255;
    float* h    = ws + o; o += (size_t)NNODES * HD;
    float* hm   = ws + o; o += (size_t)NNODES * HD;
    float* hn   = ws + o; o += (size_t)NNODES * HD;
    float* sums = ws + o; o += 256;
    float* gsum = ws + o; o += (size_t)NGRAPH * HD;       // gcnt follows contiguously
    float* gcnt = ws + o; o += NGRAPH;

    const int nb256  = (NNODES + 255) / 256;
    const int eb256  = (NEDGES + 255) / 256;
    const int nElem4 = NNODES * 32 / 256;   // blocks for N*32-thread kernels
    const int eElem4 = NEDGES * 32 / 256;   // blocks for E*32-thread kernels

    // 1) symmetric GCN normalization
    k_init_deg<<<nb256, 256, 0, stream>>>(dinv);
    k_edge_deg<<<eb256, 256, 0, stream>>>(ecol, dinv);
    k_rsqrt  <<<nb256, 256, 0, stream>>>(dinv);

    // 2) h = x @ W_in + b_in
    k_gemm_wmma<<<NNODES / 64, 256, 0, stream>>>(x, W_in, b_in, nullptr, h);

    // 3) GCN layers
    for (int i = 0; i < NLAYERS; ++i) {
        k_gemm_wmma<<<NNODES / 64, 256, 0, stream>>>(h, W_conv + (size_t)i * HD * HD,
                                                     nullptr, nullptr, hm);
        k_msg_init<<<nElem4, 256, 0, stream>>>((const float4*)hm, dinv,
                                               b_conv + (size_t)i * HD, (float4*)hn);
        k_scatter <<<eElem4, 256, 0, stream>>>((const float4*)hm, erow, ecol, dinv, hn);
        k_zero    <<<1, 256, 0, stream>>>(sums, 256);
        k_bn_stats<<<512, 256, 0, stream>>>(hn, sums);
        k_bn_apply<<<nElem4, 256, 0, stream>>>((const float4*)hn, sums,
                                               gamma + (size_t)i * HD,
                                               beta + (size_t)i * HD, (float4*)h);
    }

    // 4) global mean pool + output projection
    k_zero<<<(NGRAPH * HD + NGRAPH + 255) / 256, 256, 0, stream>>>(gsum, NGRAPH * HD + NGRAPH);
    k_pool<<<nElem4, 256, 0, stream>>>((const float4*)h, batch, gsum, gcnt);
    k_invcnt<<<(NGRAPH + 255) / 256, 256, 0, stream>>>(gcnt);
    k_gemm_wmma<<<NGRAPH / 64, 256, 0, stream>>>(gsum, W_out, b_out, gcnt, out);
}